// LinearCDE_19232863551938
// MI455X (gfx1250) — compile-verified
//
#include <hip/hip_runtime.h>

// LinearCDE on MI455X (gfx1250).
// Phase 1: A_t = Wa @ inp_t for all (b,t) via v_wmma_f32_16x16x4_f32 (4x M-blocked),
//          chunked over time so the live A chunk stays L2-resident.
// Phase 2: serial scan, one workgroup per batch; A_t double-buffered in LDS via
//          global_load_async_to_lds_b128 (ASYNCcnt) to overlap the serial chain.

typedef __attribute__((ext_vector_type(2))) float v2f;
typedef __attribute__((ext_vector_type(8))) float v8f;

#define NTOT_PAD 16384   // 8*2047 = 16376 columns, padded to 16384
#define KPAD 68          // D+1 = 65, padded to 17*4

// ---------- prep kernels: build pair-interleaved transposed operands ----------
// Layout: P[((k>>1)*ld + row)*2 + (k&1)]  -> one lane's (k0,k0+1) pair is a b64.

__global__ void cde_build_WaP(const float* __restrict__ Wa, float* __restrict__ WaP) {
    int idx = blockIdx.x * blockDim.x + threadIdx.x;      // over KPAD*16384
    if (idx >= KPAD * 16384) return;
    int j  = idx & 1;
    int pr = idx >> 1;
    int p  = pr >> 14;            // pair index
    int r  = pr & 16383;
    int k  = 2 * p + j;
    WaP[idx] = (k < 65) ? Wa[(size_t)r * 65 + k] : 0.0f;
}

__global__ void cde_build_WbP(const float* __restrict__ Wb, float* __restrict__ WbP) {
    int idx = blockIdx.x * blockDim.x + threadIdx.x;      // over KPAD*128
    if (idx >= KPAD * 128) return;
    int j  = idx & 1;
    int pr = idx >> 1;
    int p  = pr >> 7;
    int i  = pr & 127;
    int k  = 2 * p + j;
    WbP[idx] = (k < 65) ? Wb[(size_t)i * 65 + k] : 0.0f;
}

// inpP pair-interleaved over columns col = b*2047 + t, value = inp[b][t+1][k]
__global__ void cde_build_inpP(const float* __restrict__ X, float* __restrict__ inpP) {
    int idx = blockIdx.x * blockDim.x + threadIdx.x;      // over KPAD*NTOT_PAD
    if (idx >= KPAD * NTOT_PAD) return;
    int j    = idx & 1;
    int pr   = idx >> 1;
    int p    = pr >> 14;
    int colp = pr & 16383;
    int k    = 2 * p + j;
    const float dt = 1.0f / 2047.0f;
    float v = 0.0f;
    if (colp < 8 * 2047 && k < 65) {
        int b = colp / 2047;
        int t = colp % 2047;
        if (k == 0) v = dt;
        else        v = X[((size_t)b * 2048 + (t + 1)) * 64 + (k - 1)] * dt;
    }
    inpP[idx] = v;
}

// y0 = X[:,0] @ Wi.T + bi ; write to out[:,0,:] and the running-state buffer
__global__ void cde_y0(const float* __restrict__ X, const float* __restrict__ Wi,
                       const float* __restrict__ bi, float* __restrict__ out,
                       float* __restrict__ ycur) {
    int b = blockIdx.x;
    int i = threadIdx.x;                                   // 128 threads
    const float* x = X + (size_t)b * 2048 * 64;
    const float* w = Wi + (size_t)i * 64;
    float s = bi[i];
#pragma unroll 16
    for (int k = 0; k < 64; ++k) s += x[k] * w[k];
    out[((size_t)b * 2048) * 128 + i] = s;
    ycur[b * 128 + i] = s;
}

// ---------- WMMA GEMM: Out[n][r] = sum_k Asrc[r][k] * inp[col(n)][k] ----------
// Operands pair-interleaved. Each wave: 4 M-tiles (64 rows) x 1 N-tile, sharing B-frags.
// C > 0: chunk mapping n -> (b = n/C, t = t0 + n%C), col = b*2047 + min(t,2046)
// C == 0: identity (Bs run with M=128).
__global__ void __launch_bounds__(256)
cde_gemm_wmma(const float* __restrict__ AP, const float* __restrict__ BP,
              float* __restrict__ Out, int M, int Ntiles, int C, int t0) {
    int wave = (int)((blockIdx.x * blockDim.x + threadIdx.x) >> 5);
    int lane = threadIdx.x & 31;
    int Msuper = M >> 6;                        // 64 rows per wave
    int total = Msuper * Ntiles;
    if (wave >= total) return;                  // wave-uniform: EXEC all-ones for WMMA
    int msup  = wave % Msuper;
    int ntile = wave / Msuper;
    int ln = lane & 15;
    int hi = lane >> 4;

    int r0   = (msup << 6) + ln;                // first tile's A row for this lane
    int nloc = (ntile << 4) + ln;               // output column
    int col;
    if (C > 0) {
        int b = nloc / C;
        int t = t0 + (nloc % C);
        if (t > 2046) t = 2046;                 // clamped cols past end are never read
        col = b * 2047 + t;
    } else {
        col = nloc;
    }

    v8f acc0 = {0.f,0.f,0.f,0.f,0.f,0.f,0.f,0.f};
    v8f acc1 = acc0, acc2 = acc0, acc3 = acc0;
#pragma unroll
    for (int kk = 0; kk < KPAD; kk += 4) {
        int kp = (kk >> 1) + hi;                // pair index of k0 = kk + 2*hi
        const float2* ap = (const float2*)(AP + ((size_t)kp * M + r0) * 2);
        float2 bp = *(const float2*)(BP + ((size_t)kp * NTOT_PAD + col) * 2);
        v2f bf; bf.x = bp.x; bf.y = bp.y;
        float2 a0 = ap[0], a1 = ap[16], a2 = ap[32], a3 = ap[48];
        v2f af;
        af.x = a0.x; af.y = a0.y;
        acc0 = __builtin_amdgcn_wmma_f32_16x16x4_f32(false, af, false, bf, (short)0, acc0, false, false);
        af.x = a1.x; af.y = a1.y;
        acc1 = __builtin_amdgcn_wmma_f32_16x16x4_f32(false, af, false, bf, (short)0, acc1, false, false);
        af.x = a2.x; af.y = a2.y;
        acc2 = __builtin_amdgcn_wmma_f32_16x16x4_f32(false, af, false, bf, (short)0, acc2, false, false);
        af.x = a3.x; af.y = a3.y;
        acc3 = __builtin_amdgcn_wmma_f32_16x16x4_f32(false, af, false, bf, (short)0, acc3, false, false);
    }
    // D layout: VGPR v -> row v + 8*hi of the tile, col = ln. 8 consecutive rows per lane.
    size_t basen = (size_t)nloc * (size_t)M + (size_t)(msup << 6) + (size_t)(hi << 3);
    v8f* accs[4] = {&acc0, &acc1, &acc2, &acc3};
#pragma unroll
    for (int u = 0; u < 4; ++u) {
        v8f a = *accs[u];
        float4 lo, hi4;
        lo.x = a[0]; lo.y = a[1]; lo.z = a[2]; lo.w = a[3];
        hi4.x = a[4]; hi4.y = a[5]; hi4.z = a[6]; hi4.w = a[7];
        float* dst = Out + basen + (size_t)(u << 4);
        *(float4*)(dst)     = lo;
        *(float4*)(dst + 4) = hi4;
    }
}

// ---------- serial scan: one WG per batch, async double-buffered A in LDS ----------
__global__ void cde_scan(const float* __restrict__ Achunk, const float* __restrict__ Bs,
                         float* __restrict__ ycur, float* __restrict__ out,
                         int C, int t0) {
    extern __shared__ float smem[];             // [2][128*128] A buffers + y[128]
    float* buf0 = smem;
    float* buf1 = smem + 16384;
    float* ly   = smem + 32768;
    int b = blockIdx.x;
    int i = threadIdx.x;                        // 128 threads
    int steps = C;
    if (t0 + steps > 2047) steps = 2047 - t0;

    ly[i] = ycur[b * 128 + i];
    const unsigned long long Abase =
        (unsigned long long)(Achunk + (size_t)b * (size_t)C * 16384);

    // Preload step 0: thread i streams its own 512B row into buf0 (ASYNCcnt).
    {
        unsigned lds = (unsigned)(unsigned long long)(buf0 + i * 128);
        unsigned long long g = Abase + (size_t)i * 512;
#pragma unroll
        for (int q = 0; q < 32; ++q)
            asm volatile("global_load_async_to_lds_b128 %0, %1, off"
                         :: "v"(lds + q * 16), "v"(g + q * 16) : "memory");
    }
    __syncthreads();                            // ly visible

    for (int tl = 0; tl < steps; ++tl) {
        float* cur = (tl & 1) ? buf1 : buf0;
        float* nxt = (tl & 1) ? buf0 : buf1;
        if (tl + 1 < steps) {
            // issue next step's async loads, then wait for current buffer only
            unsigned lds = (unsigned)(unsigned long long)(nxt + i * 128);
            unsigned long long g = Abase + (size_t)(tl + 1) * 65536 + (size_t)i * 512;
#pragma unroll
            for (int q = 0; q < 32; ++q)
                asm volatile("global_load_async_to_lds_b128 %0, %1, off"
                             :: "v"(lds + q * 16), "v"(g + q * 16) : "memory");
            asm volatile("s_wait_asynccnt 32" ::: "memory");
        } else {
            asm volatile("s_wait_asynccnt 0" ::: "memory");
        }
        const float4* A4 = (const float4*)(cur + i * 128);
        float z = 0.0f;
#pragma unroll 8
        for (int j4 = 0; j4 < 32; ++j4) {
            float4 a = A4[j4];
            z += a.x * ly[4 * j4 + 0] + a.y * ly[4 * j4 + 1]
               + a.z * ly[4 * j4 + 2] + a.w * ly[4 * j4 + 3];
        }
        int t = t0 + tl;
        float yn = ly[i] + z + Bs[((size_t)(b * 2047 + t)) * 128 + i];
        out[((size_t)b * 2048 + t + 1) * 128 + i] = yn;
        __syncthreads();
        ly[i] = yn;
        __syncthreads();
    }
    ycur[b * 128 + i] = ly[i];
}

// ---------- host launcher ----------
extern "C" void kernel_launch(void* const* d_in, const int* in_sizes, int n_in,
                              void* d_out, int out_size, void* d_ws, size_t ws_size,
                              hipStream_t stream) {
    const float* X  = (const float*)d_in[0];
    const float* Wi = (const float*)d_in[1];
    const float* bi = (const float*)d_in[2];
    const float* Wa = (const float*)d_in[3];
    const float* Wb = (const float*)d_in[4];
    float* out = (float*)d_out;
    char* ws = (char*)d_ws;

    const size_t off_WaP  = 0;                                   // 68*16384*4 = 4,456,448
    const size_t off_WbP  = 4456448;                             // 68*128*4   =    34,816
    const size_t off_inpP = 4491264;                             // 68*16384*4 = 4,456,448
    const size_t off_Bs   = 8947712;                             // 16384*128*4= 8,388,608
    const size_t off_y    = 17336320;                            // 8*128*4    =     4,096
    const size_t off_A    = 17340416;
    float* WaP    = (float*)(ws + off_WaP);
    float* WbP    = (float*)(ws + off_WbP);
    float* inpP   = (float*)(ws + off_inpP);
    float* Bsbuf  = (float*)(ws + off_Bs);
    float* ycur   = (float*)(ws + off_y);
    float* Achunk = (float*)(ws + off_A);

    // Chunk size: multiple of 16 steps; cap at 128 so chunk (64MB) + fixed (17MB) is L2-resident.
    int C = 16;
    if (ws_size > off_A) {
        size_t steps = (ws_size - off_A) / (8ull * 16384ull * 4ull);
        steps = (steps / 16) * 16;
        if (steps > 128) steps = 128;
        if (steps >= 16) C = (int)steps;
    }

    cde_build_WaP <<<(KPAD * 16384 + 255) / 256, 256, 0, stream>>>(Wa, WaP);
    cde_build_WbP <<<(KPAD * 128 + 255) / 256, 256, 0, stream>>>(Wb, WbP);
    cde_build_inpP<<<(KPAD * NTOT_PAD + 255) / 256, 256, 0, stream>>>(X, inpP);
    cde_y0        <<<8, 128, 0, stream>>>(X, Wi, bi, out, ycur);

    { // Bs[n][i] = Wb @ inp_t  (M=128, identity column mapping over 16384 padded cols)
        int Ntiles = NTOT_PAD / 16;              // 1024
        int total  = (128 >> 6) * Ntiles;        // 2048 waves
        cde_gemm_wmma<<<(total * 32 + 255) / 256, 256, 0, stream>>>(
            WbP, inpP, Bsbuf, 128, Ntiles, 0, 0);
    }

    const int scanSmem = (2 * 16384 + 128) * 4;  // 131,584 B of the 320KB WGP LDS
    for (int t0 = 0; t0 < 2047; t0 += C) {
        int Ntiles = (8 * C) / 16;
        int total  = (16384 >> 6) * Ntiles;
        cde_gemm_wmma<<<(total * 32 + 255) / 256, 256, 0, stream>>>(
            WaP, inpP, Achunk, 16384, Ntiles, C, t0);
        cde_scan<<<8, 128, scanSmem, stream>>>(Achunk, Bsbuf, ycur, out, C, t0);
    }
}